// EncodecEuclideanCodebook_75067438399886
// MI455X (gfx1250) — compile-verified
//
#include <hip/hip_runtime.h>
#include <stdint.h>

typedef __attribute__((ext_vector_type(2))) float v2f;
typedef __attribute__((ext_vector_type(4))) float v4f;
typedef __attribute__((ext_vector_type(8))) float v8f;

typedef __attribute__((ext_vector_type(4))) unsigned int u32x4;
typedef __attribute__((ext_vector_type(4))) int          i32x4;
typedef __attribute__((ext_vector_type(8))) int          i32x8;

#define DIM 128
#define KSTEPS (DIM / 4)                 // 32 wmma k-steps of K=4
#define MTILE_WAVE 32                    // 2 x 16-row WMMA tiles per wave
#define WAVES_PER_BLOCK 8
#define MBLOCK (MTILE_WAVE * WAVES_PER_BLOCK)   // 256 rows per block
#define NTILE 32                         // bins per TDM transfer (2 subchunks)
#define ROW_DW (DIM + 2)                 // 130 dwords: TDM pad 2 dw/row (bank skew)
#define BUF_DW (NTILE * ROW_DW)          // 4160 dwords = 16640 B per buffer

// ---------------------------------------------------------------------------
// TDM issue: load one 32x128 f32 tile of E (row-major, stride 128) into LDS
// with 2-dword padding after every 128 dwords. D# built per CDNA5 ISA §8.3-8.6.
// 6-arg builtin form (clang-23 / therock): (g0, g1, g2, g3, <extra v8i>, cpol).
// ---------------------------------------------------------------------------
__device__ __forceinline__ void tdm_load_chunk(const float* gsrc, uint32_t lds_byte_addr)
{
    uint64_t ga = (uint64_t)(uintptr_t)gsrc;

    u32x4 g0;
    g0[0] = 1u;                                   // count=1 (valid), user mode
    g0[1] = lds_byte_addr;                        // lds_addr
    g0[2] = (uint32_t)(ga & 0xFFFFFFFFu);         // global_addr[31:0]
    g0[3] = (uint32_t)((ga >> 32) & 0x1FFFFFFu)   // global_addr[56:32]
          | (2u << 30);                           // type=2 ("image")

    i32x8 g1;
    g1[0] = (int)((2u << 16)                      // data_size = 4B
                | (1u << 20)                      // pad_enable
                | (6u << 22)                      // pad_interval: 128 dwords
                | (1u << 25));                    // pad_amount: 2 dwords
    g1[1] = (int)((uint32_t)DIM << 16);           // tensor_dim0[15:0] = 128
    g1[2] = (int)((uint32_t)NTILE << 16);         // tensor_dim1[15:0] = 32
    g1[3] = (int)((uint32_t)DIM << 16);           // tile_dim0 = 128
    g1[4] = NTILE;                                // tile_dim1 = 32, tile_dim2 = 0
    g1[5] = DIM;                                  // tensor_dim0_stride = 128
    g1[6] = 0;
    g1[7] = 0;

    i32x4 g2 = {0, 0, 0, 0};                      // 2D: dims 2/3 unused
    i32x4 g3 = {0, 0, 0, 0};
    i32x8 gz = {0, 0, 0, 0, 0, 0, 0, 0};          // extra arg (6-arg form), zero

    __builtin_amdgcn_tensor_load_to_lds(g0, g1, g2, g3, gz, 0);
}

// ---------------------------------------------------------------------------
// Pre-pass: esq[b] = ||embed[b]||^2  (1024 bins -> d_ws)
// ---------------------------------------------------------------------------
__global__ __launch_bounds__(256) void vq_esq_kernel(
    const float* __restrict__ embed, float* __restrict__ esq, int bins)
{
    int b = blockIdx.x * blockDim.x + threadIdx.x;
    if (b >= bins) return;
    const v4f* p = (const v4f*)(embed + (size_t)b * DIM);
    float s = 0.0f;
#pragma unroll
    for (int i = 0; i < DIM / 4; ++i) {
        v4f v = p[i];
        s += v.x * v.x + v.y * v.y + v.z * v.z + v.w * v.w;
    }
    esq[b] = s;
}

// ---------------------------------------------------------------------------
// Encode: two 16x128 A-tiles resident in VGPRs per wave (each LDS B-fragment
// feeds 2 WMMAs -> half the LDS bytes per WMMA); E staged by TDM into
// double-buffered LDS, 32 bins per transfer; 64x v_wmma_f32_16x16x4_f32 per
// 16-bin subchunk per wave.  score = 2*(x.e) - ||e||^2 ; running argmax.
// ---------------------------------------------------------------------------
__global__ __launch_bounds__(256) void vq_encode_kernel(
    const float* __restrict__ x,
    const float* __restrict__ embed,
    const float* __restrict__ esq_tab,
    float* __restrict__ idx_out,
    int nrows, int bins)
{
    __shared__ float smem[2][BUF_DW];

    const int lane = threadIdx.x & 31;
    const int wave = threadIdx.x >> 5;
    const int half = lane >> 4;          // K+2 selector (ISA 16x4 f32 layout)
    const int l15  = lane & 15;          // A: M row / B,C: N column

    const int m0 = blockIdx.x * MBLOCK + wave * MTILE_WAVE;

    // --- A fragments, loaded once: two 16-row tiles ---
    v2f a0[KSTEPS], a1[KSTEPS];
    const float* aptr0 = x + (size_t)(m0 + l15)      * DIM + 2 * half;
    const float* aptr1 = x + (size_t)(m0 + 16 + l15) * DIM + 2 * half;
#pragma unroll
    for (int ks = 0; ks < KSTEPS; ++ks) {
        a0[ks] = *(const v2f*)(aptr0 + 4 * ks);
        a1[ks] = *(const v2f*)(aptr1 + 4 * ks);
    }

    float bestVal0[8], bestVal1[8];
    int   bestIdx0[8], bestIdx1[8];
#pragma unroll
    for (int r = 0; r < 8; ++r) {
        bestVal0[r] = -3.4e38f; bestIdx0[r] = 0;
        bestVal1[r] = -3.4e38f; bestIdx1[r] = 0;
    }

    const int nbuf = bins / NTILE;       // TDM transfers (32 bins each)

    if (wave == 0)
        tdm_load_chunk(embed, (uint32_t)(uintptr_t)&smem[0][0]);

    for (int it = 0; it < nbuf; ++it) {
        const int buf = it & 1;

        if (wave == 0) {
            if (it + 1 < nbuf) {
                tdm_load_chunk(embed + (size_t)(it + 1) * NTILE * DIM,
                               (uint32_t)(uintptr_t)&smem[buf ^ 1][0]);
                __builtin_amdgcn_s_wait_tensorcnt(1);
            } else {
                __builtin_amdgcn_s_wait_tensorcnt(0);
            }
        }
        __syncthreads();                 // current transfer visible to all waves

#pragma unroll
        for (int sub = 0; sub < 2; ++sub) {
            const int n = it * NTILE + sub * 16 + l15;   // this lane's bin
            const float esq = esq_tab[n];

            const float* bbase = &smem[buf][0]
                               + (sub * 16 + l15) * ROW_DW + 2 * half;

            v8f c0a = {}, c0b = {};      // tile 0 accumulators
            v8f c1a = {}, c1b = {};      // tile 1 accumulators
#pragma unroll
            for (int ks = 0; ks < KSTEPS; ks += 2) {
                v2f b0 = *(const v2f*)(bbase + 4 * ks);
                v2f b1 = *(const v2f*)(bbase + 4 * (ks + 1));
                c0a = __builtin_amdgcn_wmma_f32_16x16x4_f32(
                          false, a0[ks],     false, b0, (short)0, c0a, false, false);
                c1a = __builtin_amdgcn_wmma_f32_16x16x4_f32(
                          false, a1[ks],     false, b0, (short)0, c1a, false, false);
                c0b = __builtin_amdgcn_wmma_f32_16x16x4_f32(
                          false, a0[ks + 1], false, b1, (short)0, c0b, false, false);
                c1b = __builtin_amdgcn_wmma_f32_16x16x4_f32(
                          false, a1[ks + 1], false, b1, (short)0, c1b, false, false);
            }

#pragma unroll
            for (int r = 0; r < 8; ++r) {
                float s0 = 2.0f * (c0a[r] + c0b[r]) - esq;
                float s1 = 2.0f * (c1a[r] + c1b[r]) - esq;
                if (s0 > bestVal0[r]) { bestVal0[r] = s0; bestIdx0[r] = n; }
                if (s1 > bestVal1[r]) { bestVal1[r] = s1; bestIdx1[r] = n; }
            }
        }

        __syncthreads();                 // protect buffer before TDM reuse
    }

    // --- argmax reduction across the 16 lanes of each half-wave -------------
    // C layout: lanes 0-15 hold rows m+r, lanes 16-31 hold rows m+r+8.
#pragma unroll
    for (int mask = 8; mask >= 1; mask >>= 1) {
#pragma unroll
        for (int r = 0; r < 8; ++r) {
            float ov0 = __shfl_xor(bestVal0[r], mask, 32);
            int   oi0 = __shfl_xor(bestIdx0[r], mask, 32);
            if (ov0 > bestVal0[r] || (ov0 == bestVal0[r] && oi0 < bestIdx0[r])) {
                bestVal0[r] = ov0; bestIdx0[r] = oi0;
            }
            float ov1 = __shfl_xor(bestVal1[r], mask, 32);
            int   oi1 = __shfl_xor(bestIdx1[r], mask, 32);
            if (ov1 > bestVal1[r] || (ov1 == bestVal1[r] && oi1 < bestIdx1[r])) {
                bestVal1[r] = ov1; bestIdx1[r] = oi1;
            }
        }
    }

    if (l15 == 0) {
#pragma unroll
        for (int r = 0; r < 8; ++r) {
            int row0 = m0 + r + half * 8;
            int row1 = m0 + 16 + r + half * 8;
            if (row0 < nrows) idx_out[row0] = (float)bestIdx0[r];
            if (row1 < nrows) idx_out[row1] = (float)bestIdx1[r];
        }
    }
}

// ---------------------------------------------------------------------------
// Decode: quantize[row] = embed[idx[row]]  (134 MB stream, float4 vectorized)
// ---------------------------------------------------------------------------
__global__ __launch_bounds__(256) void vq_gather_kernel(
    const float* __restrict__ embed,
    const float* __restrict__ idx_f,
    float* __restrict__ out,
    int nrows)
{
    int t = blockIdx.x * blockDim.x + threadIdx.x;
    int total = nrows * (DIM / 4);
    if (t >= total) return;
    int row = t >> 5;                    // 32 float4 per row
    int k4  = t & 31;
    int idx = (int)idx_f[row];
    v4f v = *(const v4f*)(embed + (size_t)idx * DIM + 4 * k4);
    *(v4f*)(out + (size_t)row * DIM + 4 * k4) = v;
}

// ---------------------------------------------------------------------------
extern "C" void kernel_launch(void* const* d_in, const int* in_sizes, int n_in,
                              void* d_out, int out_size, void* d_ws, size_t ws_size,
                              hipStream_t stream)
{
    (void)n_in; (void)out_size; (void)ws_size;
    const float* x     = (const float*)d_in[0];   // [B,T,128] fp32
    const float* embed = (const float*)d_in[1];   // [1024,128] fp32

    const int nrows = in_sizes[0] / DIM;          // 262144
    const int bins  = in_sizes[1] / DIM;          // 1024

    float* quant   = (float*)d_out;                       // [nrows,128]
    float* idx_out = quant + (size_t)nrows * DIM;         // [nrows] (as float)
    float* esq     = (float*)d_ws;                        // [bins] scratch

    vq_esq_kernel<<<dim3((bins + 255) / 256), dim3(256), 0, stream>>>(
        embed, esq, bins);

    vq_encode_kernel<<<dim3(nrows / MBLOCK), dim3(256), 0, stream>>>(
        x, embed, esq, idx_out, nrows, bins);

    int total = nrows * (DIM / 4);
    vq_gather_kernel<<<dim3((total + 255) / 256), dim3(256), 0, stream>>>(
        embed, idx_out, quant, nrows);
}